// RGCN_59107339928270
// MI455X (gfx1250) — compile-verified
//
#include <hip/hip_runtime.h>

typedef __attribute__((ext_vector_type(2))) float v2f;
typedef __attribute__((ext_vector_type(8))) float v8f;

// ---------------------------------------------------------------------------
// Degree accumulation: rsq[(2p)*N + src] += 1, rsq[(2p+1)*N + dst] += 1
// for each (layer,relation) pair p in [0,6). edges laid out [2][3][2][E].
// ---------------------------------------------------------------------------
__global__ void rgcn_degree(const int* __restrict__ edges, float* __restrict__ rsq,
                            int E, int N) {
  long idx = (long)blockIdx.x * blockDim.x + threadIdx.x;
  if (idx >= 6L * E) return;
  int p = (int)(idx / E);
  int e = (int)(idx % E);
  int s = edges[(size_t)(p * 2) * E + e];
  int d = edges[(size_t)(p * 2 + 1) * E + e];
  atomicAdd(&rsq[(size_t)(2 * p) * N + s], 1.0f);
  atomicAdd(&rsq[(size_t)(2 * p + 1) * N + d], 1.0f);
}

// deg -> rsqrt(max(deg, 1))  (DGL clamps degree to >= 1)
__global__ void rgcn_rsqrt(float* __restrict__ rsq, long n) {
  long i = (long)blockIdx.x * blockDim.x + threadIdx.x;
  if (i >= n) return;
  float v = rsq[i];
  rsq[i] = rsqrtf(fmaxf(v, 1.0f));
}

// ---------------------------------------------------------------------------
// Y[N x HOUT] = (diag(sOut) * X[N x 128]) @ W[128 x HOUT]
// One wave per 16-row strip of X; sweeps all HOUT/16 column tiles keeping the
// A fragment in registers.  V_WMMA_F32_16X16X4_F32, wave32.
//   A 16x4 f32:  lanes 0-15 -> K=k0,k0+1 ; lanes 16-31 -> K=k0+2,k0+3
//   B 4x16 f32:  VGPR0 rows K=k0 (lanes 0-15) / K=k0+2 (lanes 16-31), VGPR1 +1
//   C/D 16x16:   VGPR v: M=v (lanes 0-15), M=v+8 (lanes 16-31), N = lane&15
// ---------------------------------------------------------------------------
template <int HOUT>
__global__ void rgcn_gemm_wmma(const float* __restrict__ X,
                               const float* __restrict__ sOut,
                               const float* __restrict__ W,
                               float* __restrict__ Y, int N) {
  constexpr int K  = 128;
  constexpr int TN = HOUT / 16;
  int wave = blockIdx.x * (blockDim.x >> 5) + (threadIdx.x >> 5);
  int m0 = wave << 4;
  if (m0 >= N) return;                       // wave-uniform exit
  int lane = threadIdx.x & 31;
  int half = lane >> 4;                      // 0: lanes 0-15, 1: lanes 16-31
  int l15  = lane & 15;
  int arow = m0 + l15;
  const float* xrow = X + (size_t)arow * K;
  float s = sOut[arow];

  v8f c[TN];
#pragma unroll
  for (int t = 0; t < TN; t++) c[t] = (v8f){0.f,0.f,0.f,0.f,0.f,0.f,0.f,0.f};

  for (int k0 = 0; k0 < K; k0 += 4) {
    int ka = k0 + (half << 1);
    v2f a;
    a.x = xrow[ka] * s;
    a.y = xrow[ka + 1] * s;
    const float* w0 = W + (size_t)ka * HOUT + l15;
    const float* w1 = w0 + HOUT;
#pragma unroll
    for (int t = 0; t < TN; t++) {
      v2f b;
      b.x = w0[t * 16];
      b.y = w1[t * 16];
      c[t] = __builtin_amdgcn_wmma_f32_16x16x4_f32(
          false, a, false, b, (short)0, c[t], false, false);
    }
  }

#pragma unroll
  for (int t = 0; t < TN; t++) {
    float* yp = Y + (size_t)(m0 + (half << 3)) * HOUT + t * 16 + l15;
#pragma unroll
    for (int v = 0; v < 8; v++)
      yp[(size_t)v * HOUT] = c[t][v];
  }
}

// ---------------------------------------------------------------------------
// Edge scatter: acc[dst] += Y[src] * rsqrt(deg_in[dst]).  One wave per edge,
// each lane handles F/32 contiguous floats.  acc stays L2-resident (51 MB).
// ---------------------------------------------------------------------------
template <int F>
__global__ void rgcn_scatter(const float* __restrict__ Y,
                             const int* __restrict__ src,
                             const int* __restrict__ dst,
                             const float* __restrict__ sIn,
                             float* __restrict__ acc, int E) {
  int wid = blockIdx.x * (blockDim.x >> 5) + (threadIdx.x >> 5);
  if (wid >= E) return;                       // wave-uniform
  int lane = threadIdx.x & 31;
  int s = src[wid];
  int d = dst[wid];
  float sc = sIn[d];
  constexpr int PER = F / 32;
  const float* yp = Y + (size_t)s * F + lane * PER;
  float* ap = acc + (size_t)d * F + lane * PER;
#pragma unroll
  for (int i = 0; i < PER; i++)
    atomicAdd(&ap[i], yp[i] * sc);
}

// h = relu(acc + sum_r b1[r])   (b1 is [3][128])
__global__ void rgcn_finalize1(float* __restrict__ h, const float* __restrict__ b1,
                               long n) {
  long i = (long)blockIdx.x * blockDim.x + threadIdx.x;
  if (i >= n) return;
  int f = (int)(i & 127);
  float b = b1[f] + b1[128 + f] + b1[256 + f];
  float v = h[i] + b;
  h[i] = v > 0.f ? v : 0.f;
}

// out += sum_r b2[r]   (b2 is [3][64])
__global__ void rgcn_finalize2(float* __restrict__ out, const float* __restrict__ b2,
                               long n) {
  long i = (long)blockIdx.x * blockDim.x + threadIdx.x;
  if (i >= n) return;
  int f = (int)(i & 63);
  out[i] += b2[f] + b2[64 + f] + b2[128 + f];
}

// ---------------------------------------------------------------------------
extern "C" void kernel_launch(void* const* d_in, const int* in_sizes, int n_in,
                              void* d_out, int out_size, void* d_ws, size_t ws_size,
                              hipStream_t stream) {
  const float* x  = (const float*)d_in[0];   // [N,128]
  const float* W1 = (const float*)d_in[1];   // [3,128,128]
  const float* b1 = (const float*)d_in[2];   // [3,128]
  const float* W2 = (const float*)d_in[3];   // [3,128,64]
  const float* b2 = (const float*)d_in[4];   // [3,64]
  const int* edge = (const int*)d_in[5];     // [2,3,2,E]

  const int N = in_sizes[0] / 128;
  const int E = in_sizes[5] / 12;            // 2*3*2

  // Workspace layout (floats): rsq[12*N] | ybuf[128*N] | hbuf[128*N]
  float* rsq  = (float*)d_ws;
  float* ybuf = rsq  + (size_t)12 * N;
  float* hbuf = ybuf + (size_t)128 * N;
  float* out  = (float*)d_out;               // [N,64]

  // Degrees -> rsqrt(clamped)
  hipMemsetAsync(rsq, 0, sizeof(float) * (size_t)12 * N, stream);
  {
    long tot = 6L * E;
    rgcn_degree<<<(int)((tot + 255) / 256), 256, 0, stream>>>(edge, rsq, E, N);
  }
  {
    long tot = 12L * N;
    rgcn_rsqrt<<<(int)((tot + 255) / 256), 256, 0, stream>>>(rsq, tot);
  }

  const int waves   = (N + 15) / 16;         // 16-row strips
  const int gBlocks = (waves + 3) / 4;       // 4 waves / 128-thread block
  const int sBlocks = (E + 7) / 8;           // 8 edge-waves / 256-thread block

  // ---- Layer 1: h = relu(sum_r GraphConv(x; W1[r], b1[r])) ----
  hipMemsetAsync(hbuf, 0, sizeof(float) * (size_t)128 * N, stream);
  for (int r = 0; r < 3; r++) {
    const int p = r;                                   // layer 0 pair index
    const float* so  = rsq + (size_t)(2 * p) * N;      // rsqrt(deg_out)
    const float* si  = rsq + (size_t)(2 * p + 1) * N;  // rsqrt(deg_in)
    const int* srcp  = edge + (size_t)(p * 2) * E;
    const int* dstp  = edge + (size_t)(p * 2 + 1) * E;
    rgcn_gemm_wmma<128><<<gBlocks, 128, 0, stream>>>(
        x, so, W1 + (size_t)r * 128 * 128, ybuf, N);
    rgcn_scatter<128><<<sBlocks, 256, 0, stream>>>(ybuf, srcp, dstp, si, hbuf, E);
  }
  {
    long tot = (long)N * 128;
    rgcn_finalize1<<<(int)((tot + 255) / 256), 256, 0, stream>>>(hbuf, b1, tot);
  }

  // ---- Layer 2: out = sum_r GraphConv(h; W2[r], b2[r]) ----
  hipMemsetAsync(out, 0, sizeof(float) * (size_t)64 * N, stream);
  for (int r = 0; r < 3; r++) {
    const int p = 3 + r;                               // layer 1 pair index
    const float* so  = rsq + (size_t)(2 * p) * N;
    const float* si  = rsq + (size_t)(2 * p + 1) * N;
    const int* srcp  = edge + (size_t)(p * 2) * E;
    const int* dstp  = edge + (size_t)(p * 2 + 1) * E;
    rgcn_gemm_wmma<64><<<gBlocks, 128, 0, stream>>>(
        hbuf, so, W2 + (size_t)r * 128 * 64, ybuf, N);
    rgcn_scatter<64><<<sBlocks, 256, 0, stream>>>(ybuf, srcp, dstp, si, out, E);
  }
  {
    long tot = (long)N * 64;
    rgcn_finalize2<<<(int)((tot + 255) / 256), 256, 0, stream>>>(out, b2, tot);
  }
}